// Magwarp_68874095559122
// MI455X (gfx1250) — compile-verified
//
#include <hip/hip_runtime.h>

typedef __attribute__((ext_vector_type(2))) float v2f;
typedef __attribute__((ext_vector_type(8))) float v8f;

#define T_LEN 4096
#define C_CH  32
#define B_SZ  128
#define NK    6          // KNOT + 2 spline knots
#define P_THR 0.5f

// ---------------------------------------------------------------------------
// Kernel 1: build the (6 x 4096) transposed not-a-knot spline basis Bk[k][t]
// in workspace. One block; thread 0 solves the 6x6 system, everyone evaluates.
// ---------------------------------------------------------------------------
__global__ void spline_basis_kernel(float* __restrict__ Bk) {
  __shared__ float sS2[NK][NK];
  __shared__ float sxs[NK];

  if (threadIdx.x == 0) {
    const float step = (float)(T_LEN - 1) / (float)(NK - 1);   // 819.0
    float xs[NK], h[NK - 1];
    for (int k = 0; k < NK; ++k) xs[k] = (float)k * step;
    for (int k = 0; k < NK - 1; ++k) h[k] = xs[k + 1] - xs[k];

    // Augmented [A | R], 6 x 12
    float M[NK][2 * NK];
    for (int i = 0; i < NK; ++i)
      for (int j = 0; j < 2 * NK; ++j) M[i][j] = 0.0f;

    for (int i = 1; i < NK - 1; ++i) {
      M[i][i - 1] = h[i - 1];
      M[i][i]     = 2.0f * (h[i - 1] + h[i]);
      M[i][i + 1] = h[i];
      M[i][NK + i - 1] = 6.0f / h[i - 1];
      M[i][NK + i]     = -6.0f / h[i - 1] - 6.0f / h[i];
      M[i][NK + i + 1] = 6.0f / h[i];
    }
    // not-a-knot rows (R rows stay zero)
    M[0][0] = h[1];  M[0][1] = -(h[0] + h[1]);  M[0][2] = h[0];
    M[NK - 1][NK - 3] = h[NK - 2];
    M[NK - 1][NK - 2] = -(h[NK - 3] + h[NK - 2]);
    M[NK - 1][NK - 1] = h[NK - 3];

    // Gauss-Jordan with partial pivoting
    for (int col = 0; col < NK; ++col) {
      int p = col;
      for (int r = col + 1; r < NK; ++r)
        if (fabsf(M[r][col]) > fabsf(M[p][col])) p = r;
      if (p != col)
        for (int j = 0; j < 2 * NK; ++j) {
          float t = M[col][j]; M[col][j] = M[p][j]; M[p][j] = t;
        }
      float inv = 1.0f / M[col][col];
      for (int j = col; j < 2 * NK; ++j) M[col][j] *= inv;
      for (int r = 0; r < NK; ++r) {
        if (r == col) continue;
        float f = M[r][col];
        if (f != 0.0f)
          for (int j = col; j < 2 * NK; ++j) M[r][j] -= f * M[col][j];
      }
    }
    for (int i = 0; i < NK; ++i) {
      for (int k = 0; k < NK; ++k) sS2[i][k] = M[i][NK + k];
      sxs[i] = xs[i];
    }
  }
  __syncthreads();

  for (int t = threadIdx.x; t < T_LEN; t += blockDim.x) {
    const float te = (float)t;
    int idx = 0;
    for (int j = 1; j <= NK - 2; ++j)
      if (te >= sxs[j]) idx = j;               // == clip(searchsorted-1, 0, n-2)
    const float hh   = sxs[idx + 1] - sxs[idx];
    const float dl   = te - sxs[idx];
    const float dr   = sxs[idx + 1] - te;
    const float dl3  = dl * dl * dl;
    const float dr3  = dr * dr * dr;
    const float c6   = 1.0f / (6.0f * hh);
    const float h26  = hh * hh / 6.0f;
    const float invh = 1.0f / hh;
#pragma unroll
    for (int k = 0; k < NK; ++k) {
      const float Si = sS2[idx][k];
      const float Sj = sS2[idx + 1][k];
      const float Yi = (idx == k)     ? 1.0f : 0.0f;
      const float Yj = (idx + 1 == k) ? 1.0f : 0.0f;
      const float w = (Si * dr3 + Sj * dl3) * c6
                    + (Yi - Si * h26) * dr * invh
                    + (Yj - Sj * h26) * dl * invh;
      Bk[k * T_LEN + t] = w;                   // transposed: column-major in t
    }
  }
}

// ---------------------------------------------------------------------------
// Kernel 2: per wave: one batch b, one 16-channel half, strided 16-wide t
// tiles. curves tile (16c x 16t) = yy_tileᵀ (16x6) @ Bmat_tileᵀ (6x16) via two
// V_WMMA_F32_16X16X4_F32 (K = 0..3 and K = 4..5 with upper lanes zeroed).
// ---------------------------------------------------------------------------
__global__ void magwarp_wmma_kernel(const float* __restrict__ x,
                                    const float* __restrict__ yy,
                                    const float* __restrict__ mask_u,
                                    const float* __restrict__ Bk,
                                    float* __restrict__ out) {
  const int lane = threadIdx.x & 31;
  const int wv   = threadIdx.x >> 5;
  const int b    = blockIdx.z;
  const int c0   = blockIdx.y * 16;
  const int m    = lane & 15;        // M (c) for A, N (t) for B/D
  const int hi   = lane >> 4;        // lane-half selects K base (0 or 2)
  const bool apply = mask_u[b] < P_THR;   // uniform per block

  // A operand: yy[b, k, c0+m]; VGPR0 = {K0 | K2}, VGPR1 = {K1 | K3}
  const float* yyb = yy + (size_t)b * NK * C_CH;
  v2f a0, a1;
  a0.x = yyb[(2 * hi + 0) * C_CH + c0 + m];
  a0.y = yyb[(2 * hi + 1) * C_CH + c0 + m];
  const float a4 = yyb[4 * C_CH + c0 + m];
  const float a5 = yyb[5 * C_CH + c0 + m];
  a1.x = hi ? 0.0f : a4;             // K=6,7 don't exist -> zero pad
  a1.y = hi ? 0.0f : a5;

  const size_t base = (size_t)b * C_CH * T_LEN;
  const int waves_total = gridDim.x * (blockDim.x >> 5);
  const int wave_id     = blockIdx.x * (blockDim.x >> 5) + wv;
  const int ntiles      = T_LEN / 16;

  for (int tile = wave_id; tile < ntiles; tile += waves_total) {
    const int t0 = tile * 16;

    // B operand: Bmat[t0+m, k]; same lane-half K striping as A
    v2f b0, b1;
    b0.x = Bk[(2 * hi + 0) * T_LEN + t0 + m];
    b0.y = Bk[(2 * hi + 1) * T_LEN + t0 + m];
    const float b4 = Bk[4 * T_LEN + t0 + m];
    const float b5 = Bk[5 * T_LEN + t0 + m];
    b1.x = hi ? 0.0f : b4;
    b1.y = hi ? 0.0f : b5;

    v8f acc = {};
    acc = __builtin_amdgcn_wmma_f32_16x16x4_f32(false, a0, false, b0,
                                                (short)0, acc, false, false);
    acc = __builtin_amdgcn_wmma_f32_16x16x4_f32(false, a1, false, b1,
                                                (short)0, acc, false, false);

    // D layout: VGPR r, lane: M = r + 8*hi (channel), N = m (time)
    const int t = t0 + m;
#pragma unroll
    for (int r = 0; r < 8; ++r) {
      const int c = c0 + r + 8 * hi;
      const size_t o = base + (size_t)c * T_LEN + t;
      const float xv = x[o];
      out[o] = apply ? xv * acc[r] : xv;
    }
  }
}

extern "C" void kernel_launch(void* const* d_in, const int* in_sizes, int n_in,
                              void* d_out, int out_size, void* d_ws, size_t ws_size,
                              hipStream_t stream) {
  const float* x_t    = (const float*)d_in[0];
  const float* yy     = (const float*)d_in[1];
  const float* mask_u = (const float*)d_in[2];
  float* out = (float*)d_out;
  float* Bk  = (float*)d_ws;                 // 6 * 4096 * 4 = 96 KB scratch

  spline_basis_kernel<<<1, 256, 0, stream>>>(Bk);

  dim3 grid(8, C_CH / 16, B_SZ);             // 2048 blocks x 8 waves
  magwarp_wmma_kernel<<<grid, 256, 0, stream>>>(x_t, yy, mask_u, Bk, out);
}